// NumReps_154618822956
// MI455X (gfx1250) — compile-verified
//
#include <hip/hip_runtime.h>

// Problem constants from the reference
#define B_    16
#define M_    128
#define S_    2048
#define D_    1024
#define PCT   0.2f
#define NT    256          // threads per block (8 wave32 waves)
#define MAXW  12           // staged window rows (actual max = 10) -> 48KB LDS tile

typedef int v4i __attribute__((ext_vector_type(4)));
typedef __attribute__((address_space(1))) v4i* gv4_t;   // global 16B vector ptr
typedef __attribute__((address_space(3))) v4i* lv4_t;   // LDS    16B vector ptr
typedef __attribute__((address_space(3))) void* la_t;

#if defined(__AMDGCN__) && defined(__has_builtin)
#  if __has_builtin(__builtin_amdgcn_global_load_async_to_lds_b128)
#    define HAVE_ASYNC_B128 1
#  else
#    define HAVE_ASYNC_B128 0
#  endif
#  if __has_builtin(__builtin_amdgcn_s_wait_asynccnt)
#    define HAVE_WAIT_ASYNC 1
#  else
#    define HAVE_WAIT_ASYNC 0
#  endif
#else
#  define HAVE_ASYNC_B128 0
#  define HAVE_WAIT_ASYNC 0
#endif

__device__ __forceinline__ void wait_asynccnt0() {
#if HAVE_WAIT_ASYNC
  __builtin_amdgcn_s_wait_asynccnt(0);
#else
  asm volatile("s_wait_asynccnt 0" ::: "memory");
#endif
}

__global__ __launch_bounds__(NT) void numreps_window_mean_kernel(
    const float* __restrict__ mask,   // [B, M, S]
    const float* __restrict__ reps,   // [B, S, D]
    float* __restrict__ out)          // [B, M, D]
{
  __shared__ int   sFirst, sLast, sStart, sCount;
  __shared__ float tile[MAXW * D_];   // 48 KB window staging

  const int row = blockIdx.x;         // 0 .. B*M-1
  const int b   = row / M_;
  const int t   = threadIdx.x;

  if (t == 0) { sFirst = 0x7fffffff; sLast = -1; }
  __syncthreads();

  // ---- Phase 1: find first/last one in this mask row (coalesced scan) ----
  const float* mrow = mask + (size_t)row * S_;
  int lf = 0x7fffffff, ll = -1;
#pragma unroll
  for (int i = 0; i < S_ / NT; ++i) {
    const int s = t + i * NT;
    if (mrow[s] > 0.5f) { lf = min(lf, s); ll = max(ll, s); }
  }
  if (ll >= 0) { atomicMin(&sFirst, lf); atomicMax(&sLast, ll); }
  __syncthreads();

  // ---- Phase 2: expand_window semantics (float mult, int truncation) ----
  if (t == 0) {
    if (sLast < 0) { sStart = 0; sCount = 0; }        // empty row -> zeros
    else {
      const int   first  = sFirst, last = sLast;
      const float window = (float)(last - first + 1);
      const int   expand = (int)(window * PCT);       // truncates like .int()
      const int   ns = max(first - expand, 0);
      const int   ne = min(last + expand, S_ - 1);
      sStart = ns; sCount = ne - ns + 1;
    }
  }
  __syncthreads();

  const int start = sStart;
  const int count = sCount;
  float4*   orow4 = (float4*)(out + (size_t)row * D_);   // D_/4 == NT

  if (count == 0) { orow4[t] = make_float4(0.f, 0.f, 0.f, 0.f); return; }

  const float* rbase = reps + (size_t)b * S_ * D_ + (size_t)start * D_;

  if (count <= MAXW) {
    // ---- Phase 3: async-stage `count` contiguous 4KB rows into LDS ----
    // Each thread moves one 16-byte chunk per row (256 chunks * 16B = 4KB).
    for (int k = 0; k < count; ++k) {
      const int fidx = k * D_ + t * 4;                 // float index
#if HAVE_ASYNC_B128
      __builtin_amdgcn_global_load_async_to_lds_b128(
          (gv4_t)(rbase + fidx),
          (lv4_t)(&tile[fidx]),
          /*imm offset*/0, /*cpol*/0);
#else
      // fallback: attempt raw CDNA5 async-to-LDS encoding
      unsigned lds_addr = (unsigned)(unsigned long long)(la_t)(void*)(&tile[fidx]);
      unsigned voff     = (unsigned)(fidx * (int)sizeof(float));
      asm volatile("global_load_async_to_lds_b128 %0, %1, %2 offset:0"
                   :: "v"(lds_addr), "v"(voff), "s"(rbase) : "memory");
#endif
    }
    wait_asynccnt0();      // this wave's async loads complete
    __syncthreads();       // all waves' chunks visible in LDS

    // ---- Phase 4: mean over <=10 rows, one float4 column slice / thread ----
    const float4* t4 = (const float4*)tile;
    float4 acc = make_float4(0.f, 0.f, 0.f, 0.f);
    for (int k = 0; k < count; ++k) {
      const float4 v = t4[k * (D_ / 4) + t];
      acc.x += v.x; acc.y += v.y; acc.z += v.z; acc.w += v.w;
    }
    const float inv = 1.0f / (float)count;             // count>=1 == clamp(min=1)
    acc.x *= inv; acc.y *= inv; acc.z *= inv; acc.w *= inv;
    orow4[t] = acc;
  } else {
    // Robust fallback (cannot trigger with this problem's data): global accum
    float4 acc = make_float4(0.f, 0.f, 0.f, 0.f);
    for (int k = 0; k < count; ++k) {
      const float4 v = ((const float4*)(rbase + (size_t)k * D_))[t];
      acc.x += v.x; acc.y += v.y; acc.z += v.z; acc.w += v.w;
    }
    const float inv = 1.0f / (float)count;
    acc.x *= inv; acc.y *= inv; acc.z *= inv; acc.w *= inv;
    orow4[t] = acc;
  }
}

extern "C" void kernel_launch(void* const* d_in, const int* in_sizes, int n_in,
                              void* d_out, int out_size, void* d_ws, size_t ws_size,
                              hipStream_t stream) {
  (void)in_sizes; (void)n_in; (void)d_ws; (void)ws_size; (void)out_size;
  const float* number_mask = (const float*)d_in[0];   // [B, M, S] f32
  const float* reps        = (const float*)d_in[1];   // [B, S, D] f32
  float*       out         = (float*)d_out;           // [B, M, D] f32

  dim3 grid(B_ * M_);   // one block per output row
  dim3 block(NT);
  numreps_window_mean_kernel<<<grid, block, 0, stream>>>(number_mask, reps, out);
}